// GAT_781684048444
// MI455X (gfx1250) — compile-verified
//
#include <hip/hip_runtime.h>
#include <stdint.h>
#include <math.h>

// ---------------- problem constants (match reference) ----------------
#define N_NODES 10000
#define N_EDGES 160000
#define IN_F    128
#define HID     64
#define H1      8
#define OUT_F   4
#define F1      (H1 * HID)   // 512

typedef __attribute__((ext_vector_type(2))) float v2f;
typedef __attribute__((ext_vector_type(8))) float v8f;

// ---------------- helpers ----------------
__device__ __forceinline__ void atomicMaxFloat(float* addr, float val) {
  // invariant-preserving float max via int max (val>=0) / uint min (val<0)
  if (val >= 0.0f) {
    atomicMax((int*)addr, __float_as_int(val));
  } else {
    atomicMin((unsigned int*)addr, (unsigned int)__float_as_int(val));
  }
}

// ---------------- init: accumulators + per-head edge constants ----------------
__global__ void gat_init(float* __restrict__ out1, float* __restrict__ amax1,
                         float* __restrict__ denom1, float* __restrict__ amax2,
                         float* __restrict__ denom2, float* __restrict__ dout,
                         float* __restrict__ ce1, float* __restrict__ ce2,
                         const float* __restrict__ We1, const float* __restrict__ ae1,
                         const float* __restrict__ We2, const float* __restrict__ ae2) {
  int idx = blockIdx.x * blockDim.x + threadIdx.x;
  if (idx < N_NODES * F1) out1[idx] = 0.0f;
  if (idx < N_NODES * H1) { amax1[idx] = -3.0e38f; denom1[idx] = 0.0f; }
  if (idx < N_NODES)      { amax2[idx] = -3.0e38f; denom2[idx] = 0.0f; }
  if (idx < N_NODES * OUT_F) dout[idx] = 0.0f;
  if (idx < H1) {
    float s = 0.0f;
    for (int c = 0; c < HID; ++c) s += We1[idx * HID + c] * ae1[idx * HID + c];
    ce1[idx] = s;
  }
  if (idx == 0) {
    float s = 0.0f;
    for (int c = 0; c < OUT_F; ++c) s += We2[c] * ae2[c];
    ce2[0] = s;
  }
}

// ---------------- GEMM1: xh = x @ W1  (f32 WMMA 16x16x4) ----------------
// 625 M-tiles x 32 N-tiles = 20000 tiles; 1 tile per wave, 8 waves per block.
__global__ __launch_bounds__(256) void gemm1_wmma(const float* __restrict__ x,
                                                  const float* __restrict__ W1,
                                                  float* __restrict__ xh) {
  const int wave = threadIdx.x >> 5;
  const int lane = threadIdx.x & 31;
  const int tile = blockIdx.x * 8 + wave;       // exactly 20000
  const int NT   = F1 / 16;                     // 32
  const int mt   = tile / NT;
  const int nt   = tile - mt * NT;
  const int half = lane >> 4;                   // 0: K=2h+0.., 1: K=2h+2..
  const int l16  = lane & 15;
  const int row  = mt * 16 + l16;               // A row (N=10000=625*16 exact)
  const int col  = nt * 16 + l16;               // B/D column

  v8f acc = {};
  const float* xr = x  + row * IN_F + half * 2;   // A: K = k0 + 2*half + {0,1}
  const float* wc = W1 + (half * 2) * F1 + col;   // B: same K striping

#pragma unroll
  for (int k0 = 0; k0 < IN_F; k0 += 4) {
    v2f a = *(const v2f*)(xr + k0);
    v2f b;
    b.x = wc[k0 * F1];
    b.y = wc[k0 * F1 + F1];
    acc = __builtin_amdgcn_wmma_f32_16x16x4_f32(false, a, false, b,
                                                (short)0, acc, false, false);
  }
  // D layout: vgpr r, lanes<16 -> M=r ; lanes>=16 -> M=r+8 ; N = lane%16
  float* o = xh + (mt * 16 + half * 8) * F1 + col;
#pragma unroll
  for (int r = 0; r < 8; ++r) o[r * F1] = acc[r];
}

// ---------------- per-node attention logits, layer 1 ----------------
__global__ void node_alpha1(const float* __restrict__ xh,
                            const float* __restrict__ att_src,
                            const float* __restrict__ att_dst,
                            float* __restrict__ as1, float* __restrict__ ad1) {
  int idx = blockIdx.x * blockDim.x + threadIdx.x;
  if (idx >= N_NODES * H1) return;
  int n = idx >> 3, h = idx & 7;
  const float* v  = xh + n * F1 + h * HID;
  const float* ws = att_src + h * HID;
  const float* wd = att_dst + h * HID;
  float s = 0.0f, d = 0.0f;
  for (int c = 0; c < HID; ++c) { float t = v[c]; s += t * ws[c]; d += t * wd[c]; }
  as1[idx] = s; ad1[idx] = d;
}

// ---------------- edge pass A (layer1): logits + segment max ----------------
__global__ void edge1_alpha(const int* __restrict__ ei, const float* __restrict__ w,
                            const float* __restrict__ as1, const float* __restrict__ ad1,
                            const float* __restrict__ ce1,
                            float* __restrict__ alpha1, float* __restrict__ amax1) {
  int idx = blockIdx.x * blockDim.x + threadIdx.x;
  if (idx >= N_EDGES * H1) return;
  int e = idx >> 3, h = idx & 7;
  int s = ei[e], d = ei[N_EDGES + e];
  float a = as1[s * H1 + h] + ad1[d * H1 + h] + w[e] * ce1[h];
  a = (a > 0.0f) ? a : 0.2f * a;                // leaky_relu(0.2)
  alpha1[idx] = a;
  atomicMaxFloat(&amax1[d * H1 + h], a);
}

// ---------------- edge pass B (layer1): exp + segment sum ----------------
__global__ void edge1_exp(const int* __restrict__ ei,
                          float* __restrict__ alpha1,
                          const float* __restrict__ amax1,
                          float* __restrict__ denom1) {
  int idx = blockIdx.x * blockDim.x + threadIdx.x;
  if (idx >= N_EDGES * H1) return;
  int e = idx >> 3, h = idx & 7;
  int d = ei[N_EDGES + e];
  float a = expf(alpha1[idx] - amax1[d * H1 + h]);
  alpha1[idx] = a;
  atomicAdd(&denom1[d * H1 + h], a);
}

// ---------------- edge pass C (layer1): gather * alpha -> scatter-add -------
// one block per edge, 512 threads = (h,c); fully coalesced loads and atomics.
__global__ __launch_bounds__(512) void scatter1(const int* __restrict__ ei,
                                                const float* __restrict__ xh,
                                                const float* __restrict__ alpha1,
                                                const float* __restrict__ denom1,
                                                float* __restrict__ out1) {
  int e = blockIdx.x;
  int t = threadIdx.x;          // h*64 + c
  int h = t >> 6;
  int s = ei[e], d = ei[N_EDGES + e];
  float coeff = alpha1[e * H1 + h] / (denom1[d * H1 + h] + 1e-16f);
  atomicAdd(&out1[d * F1 + t], xh[s * F1 + t] * coeff);
}

// ---------------- h1 = elu(agg + b1), in place ----------------
__global__ void elu_bias(float* __restrict__ out1, const float* __restrict__ b1) {
  int idx = blockIdx.x * blockDim.x + threadIdx.x;
  if (idx >= N_NODES * F1) return;
  float v = out1[idx] + b1[idx & (F1 - 1)];
  out1[idx] = (v > 0.0f) ? v : (expf(v) - 1.0f);
}

// ---------------- GEMM2: y2 = h1 @ W2  (f32 WMMA, N padded 4->16) ----------
__global__ __launch_bounds__(256) void gemm2_wmma(const float* __restrict__ h1,
                                                  const float* __restrict__ W2,
                                                  float* __restrict__ y2) {
  const int wave = threadIdx.x >> 5;
  const int lane = threadIdx.x & 31;
  const int mt = blockIdx.x * 8 + wave;
  if (mt >= N_NODES / 16) return;               // uniform per wave
  const int half = lane >> 4, l16 = lane & 15;
  const int row = mt * 16 + l16;
  const bool live = (l16 < OUT_F);

  v8f acc = {};
  const float* hr = h1 + row * F1 + half * 2;
#pragma unroll 8
  for (int k0 = 0; k0 < F1; k0 += 4) {
    v2f a = *(const v2f*)(hr + k0);
    int k = k0 + half * 2;
    v2f b;
    b.x = live ? W2[k * OUT_F + l16] : 0.0f;
    b.y = live ? W2[(k + 1) * OUT_F + l16] : 0.0f;
    acc = __builtin_amdgcn_wmma_f32_16x16x4_f32(false, a, false, b,
                                                (short)0, acc, false, false);
  }
  if (live) {
    float* o = y2 + (mt * 16 + half * 8) * OUT_F + l16;
#pragma unroll
    for (int r = 0; r < 8; ++r) o[r * OUT_F] = acc[r];
  }
}

// ---------------- per-node attention logits, layer 2 ----------------
__global__ void node_alpha2(const float* __restrict__ y2,
                            const float* __restrict__ att_src,
                            const float* __restrict__ att_dst,
                            float* __restrict__ as2, float* __restrict__ ad2) {
  int n = blockIdx.x * blockDim.x + threadIdx.x;
  if (n >= N_NODES) return;
  const float* v = y2 + n * OUT_F;
  float s = 0.0f, d = 0.0f;
  for (int c = 0; c < OUT_F; ++c) { s += v[c] * att_src[c]; d += v[c] * att_dst[c]; }
  as2[n] = s; ad2[n] = d;
}

__global__ void edge2_alpha(const int* __restrict__ ei, const float* __restrict__ w,
                            const float* __restrict__ as2, const float* __restrict__ ad2,
                            const float* __restrict__ ce2,
                            float* __restrict__ alpha2, float* __restrict__ amax2) {
  int e = blockIdx.x * blockDim.x + threadIdx.x;
  if (e >= N_EDGES) return;
  int s = ei[e], d = ei[N_EDGES + e];
  float a = as2[s] + ad2[d] + w[e] * ce2[0];
  a = (a > 0.0f) ? a : 0.2f * a;
  alpha2[e] = a;
  atomicMaxFloat(&amax2[d], a);
}

__global__ void edge2_exp(const int* __restrict__ ei, float* __restrict__ alpha2,
                          const float* __restrict__ amax2, float* __restrict__ denom2) {
  int e = blockIdx.x * blockDim.x + threadIdx.x;
  if (e >= N_EDGES) return;
  int d = ei[N_EDGES + e];
  float a = expf(alpha2[e] - amax2[d]);
  alpha2[e] = a;
  atomicAdd(&denom2[d], a);
}

__global__ void scatter2(const int* __restrict__ ei, const float* __restrict__ y2,
                         const float* __restrict__ alpha2, const float* __restrict__ denom2,
                         float* __restrict__ dout) {
  int idx = blockIdx.x * blockDim.x + threadIdx.x;
  if (idx >= N_EDGES * OUT_F) return;
  int e = idx >> 2, c = idx & 3;
  int s = ei[e], d = ei[N_EDGES + e];
  float coeff = alpha2[e] / (denom2[d] + 1e-16f);
  atomicAdd(&dout[d * OUT_F + c], y2[s * OUT_F + c] * coeff);
}

__global__ void final_bias(float* __restrict__ dout, const float* __restrict__ b2) {
  int idx = blockIdx.x * blockDim.x + threadIdx.x;
  if (idx >= N_NODES * OUT_F) return;
  dout[idx] += b2[idx & 3];
}

// ---------------- launch ----------------
extern "C" void kernel_launch(void* const* d_in, const int* in_sizes, int n_in,
                              void* d_out, int out_size, void* d_ws, size_t ws_size,
                              hipStream_t stream) {
  const float* x   = (const float*)d_in[0];
  const int*   ei  = (const int*)d_in[1];      // edge_index (JAX default -> int32)
  const float* w   = (const float*)d_in[2];
  const float* W1  = (const float*)d_in[3];
  const float* aS1 = (const float*)d_in[4];
  const float* aD1 = (const float*)d_in[5];
  const float* aE1 = (const float*)d_in[6];
  const float* We1 = (const float*)d_in[7];
  const float* b1  = (const float*)d_in[8];
  const float* W2  = (const float*)d_in[9];
  const float* aS2 = (const float*)d_in[10];
  const float* aD2 = (const float*)d_in[11];
  const float* aE2 = (const float*)d_in[12];
  const float* We2 = (const float*)d_in[13];
  const float* b2  = (const float*)d_in[14];
  float* dout = (float*)d_out;

  float* ws = (float*)d_ws;
  size_t o = 0;
  float* xh     = ws + o; o += (size_t)N_NODES * F1;   // 5.12M
  float* out1   = ws + o; o += (size_t)N_NODES * F1;   // agg1 / h1 (in place)
  float* alpha1 = ws + o; o += (size_t)N_EDGES * H1;
  float* as1    = ws + o; o += (size_t)N_NODES * H1;
  float* ad1    = ws + o; o += (size_t)N_NODES * H1;
  float* amax1  = ws + o; o += (size_t)N_NODES * H1;
  float* denom1 = ws + o; o += (size_t)N_NODES * H1;
  float* y2     = ws + o; o += (size_t)N_NODES * OUT_F;
  float* alpha2 = ws + o; o += (size_t)N_EDGES;
  float* as2    = ws + o; o += (size_t)N_NODES;
  float* ad2    = ws + o; o += (size_t)N_NODES;
  float* amax2  = ws + o; o += (size_t)N_NODES;
  float* denom2 = ws + o; o += (size_t)N_NODES;
  float* ce1    = ws + o; o += 8;
  float* ce2    = ws + o; o += 1;
  (void)in_sizes; (void)n_in; (void)out_size; (void)ws_size; (void)o;

  const int B = 256;
  gat_init<<<(N_NODES * F1 + B - 1) / B, B, 0, stream>>>(out1, amax1, denom1, amax2,
                                                         denom2, dout, ce1, ce2,
                                                         We1, aE1, We2, aE2);
  gemm1_wmma<<<(N_NODES / 16) * (F1 / 16) / 8, B, 0, stream>>>(x, W1, xh);
  node_alpha1<<<(N_NODES * H1 + B - 1) / B, B, 0, stream>>>(xh, aS1, aD1, as1, ad1);
  edge1_alpha<<<(N_EDGES * H1 + B - 1) / B, B, 0, stream>>>(ei, w, as1, ad1, ce1,
                                                            alpha1, amax1);
  edge1_exp<<<(N_EDGES * H1 + B - 1) / B, B, 0, stream>>>(ei, alpha1, amax1, denom1);
  scatter1<<<N_EDGES, F1, 0, stream>>>(ei, xh, alpha1, denom1, out1);
  elu_bias<<<(N_NODES * F1 + B - 1) / B, B, 0, stream>>>(out1, b1);
  gemm2_wmma<<<(N_NODES / 16 + 7) / 8, B, 0, stream>>>(out1, W2, y2);
  node_alpha2<<<(N_NODES + B - 1) / B, B, 0, stream>>>(y2, aS2, aD2, as2, ad2);
  edge2_alpha<<<(N_EDGES + B - 1) / B, B, 0, stream>>>(ei, w, as2, ad2, ce2,
                                                       alpha2, amax2);
  edge2_exp<<<(N_EDGES + B - 1) / B, B, 0, stream>>>(ei, alpha2, amax2, denom2);
  scatter2<<<(N_EDGES * OUT_F + B - 1) / B, B, 0, stream>>>(ei, y2, alpha2, denom2, dout);
  final_bias<<<(N_NODES * OUT_F + B - 1) / B, B, 0, stream>>>(dout, b2);
}